// SAGEConv_73349451481846
// MI455X (gfx1250) — compile-verified
//
#include <hip/hip_runtime.h>

#define N_NODES 50000
#define FEAT    64            // in/out feature dim
#define KDIM    128           // 2*FEAT
#define LDS_STRIDE 65         // padded row stride for W in LDS (break bank conflicts)

typedef __attribute__((ext_vector_type(2))) float v2f;
typedef __attribute__((ext_vector_type(8))) float v8f;

// ---------------------------------------------------------------- zero ws ---
__global__ void sage_zero_kernel(float* __restrict__ ws, int n) {
    int i = blockIdx.x * blockDim.x + threadIdx.x;
    int stride = gridDim.x * blockDim.x;
    for (; i < n; i += stride) ws[i] = 0.0f;
}

// ------------------------------------------------------------- scatter-add ---
// One wave32 per edge: lane L loads float2 of the 64-wide source row and
// atomically accumulates into summed[dst]; lane 0 counts degree.
__global__ __launch_bounds__(256) void sage_scatter_kernel(
    const float2* __restrict__ h2,     // h viewed as [N][32] float2
    const int*    __restrict__ src,
    const int*    __restrict__ dst,
    float*        __restrict__ summed, // [N][64]
    float*        __restrict__ deg,    // [N]
    int n_edges)
{
    int edge = (blockIdx.x * blockDim.x + threadIdx.x) >> 5;
    int lane = threadIdx.x & 31;
    if (edge >= n_edges) return;
    int s = src[edge];
    int d = dst[edge];
    float2 v = h2[(size_t)s * 32 + lane];
    float* dr = summed + (size_t)d * 64 + 2 * lane;
    unsafeAtomicAdd(dr,     v.x);   // global_atomic_add_f32 (L2-resident accumulator)
    unsafeAtomicAdd(dr + 1, v.y);
    if (lane == 0) unsafeAtomicAdd(&deg[d], 1.0f);
}

// --------------------------------------------- fused mean+concat+GEMM+bias ---
// out[m][n] = sum_k htot[m][k] * W[k][n] + b[n],
// htot[m][k] = (k<64) ? h[m][k] : summed[m][k-64] / max(deg[m],1)
// Wave-per-16x16-tile using V_WMMA_F32_16X16X4_F32 (f32 precision end-to-end).
__global__ __launch_bounds__(256) void sage_gemm_kernel(
    const float* __restrict__ h,       // [N][64]
    const float* __restrict__ summed,  // [N][64]
    const float* __restrict__ deg,     // [N]
    const float* __restrict__ W,       // [128][64]
    const float* __restrict__ bias,    // [64]
    float*       __restrict__ out)     // [N][64]
{
    __shared__ float Wl[KDIM * LDS_STRIDE];

    int tid = threadIdx.x;
    // Stage W (128x64 = 32KB) into LDS, padded stride.
    for (int i = tid; i < KDIM * FEAT; i += 256) {
        int k = i >> 6, n = i & 63;
        Wl[k * LDS_STRIDE + n] = W[i];
    }
    __syncthreads();

    int wave = tid >> 5;            // 8 waves: 2 (M) x 4 (N) tiles
    int lane = tid & 31;
    int mt   = wave >> 2;           // 0..1
    int nt   = wave & 3;            // 0..3
    int mbase = blockIdx.x * 32 + mt * 16;
    int l15   = lane & 15;
    int ncol  = nt * 16 + l15;      // B/C/D: N index = lane&15 within tile
    bool hiK  = (lane >= 16);       // A/B: lanes 16-31 carry K+2,K+3 of each K=4 step

    int mrow = mbase + l15;                            // A: M = lane&15
    int mc   = (mrow < N_NODES) ? mrow : (N_NODES - 1); // clamp: EXEC must stay all-1s
    float rd = 1.0f / fmaxf(deg[mc], 1.0f);

    const float* hrow = h      + (size_t)mc * 64;
    const float* srow = summed + (size_t)mc * 64;

    v8f acc = {};
    // K = 0..63 : self features from h
    #pragma unroll
    for (int k = 0; k < 16; ++k) {
        int kb = 4 * k + (hiK ? 2 : 0);
        v2f a;  a.x = hrow[kb];                 a.y = hrow[kb + 1];
        v2f bb; bb.x = Wl[kb * LDS_STRIDE + ncol];
                bb.y = Wl[(kb + 1) * LDS_STRIDE + ncol];
        acc = __builtin_amdgcn_wmma_f32_16x16x4_f32(
                  false, a, false, bb, (short)0, acc, false, false);
    }
    // K = 64..127 : neighbor mean computed on the fly (summed * 1/deg)
    #pragma unroll
    for (int k = 0; k < 16; ++k) {
        int kb = 4 * k + (hiK ? 2 : 0);
        int kw = kb + 64;
        v2f a;  a.x = srow[kb] * rd;            a.y = srow[kb + 1] * rd;
        v2f bb; bb.x = Wl[kw * LDS_STRIDE + ncol];
                bb.y = Wl[(kw + 1) * LDS_STRIDE + ncol];
        acc = __builtin_amdgcn_wmma_f32_16x16x4_f32(
                  false, a, false, bb, (short)0, acc, false, false);
    }

    float bv = bias[ncol];
    // C/D layout: lanes 0-15 -> M = r, lanes 16-31 -> M = r + 8
    #pragma unroll
    for (int r = 0; r < 8; ++r) {
        int m = mbase + r + (hiK ? 8 : 0);
        if (m < N_NODES) out[(size_t)m * 64 + ncol] = acc[r] + bv;
    }
}

// ------------------------------------------------------------------ launch ---
extern "C" void kernel_launch(void* const* d_in, const int* in_sizes, int n_in,
                              void* d_out, int out_size, void* d_ws, size_t ws_size,
                              hipStream_t stream) {
    const float* h    = (const float*)d_in[0];
    const int*   src  = (const int*)  d_in[1];
    const int*   dst  = (const int*)  d_in[2];
    const float* W    = (const float*)d_in[3];
    const float* bias = (const float*)d_in[4];
    float*       out  = (float*)d_out;
    int n_edges = in_sizes[1];

    float* summed = (float*)d_ws;                     // N_NODES * 64 floats
    float* deg    = summed + (size_t)N_NODES * 64;    // N_NODES floats

    int zn = N_NODES * 64 + N_NODES;
    sage_zero_kernel<<<2048, 256, 0, stream>>>(summed, zn);

    int sc_blocks = (n_edges + 7) / 8;                // 8 waves/block, 1 edge/wave
    sage_scatter_kernel<<<sc_blocks, 256, 0, stream>>>(
        (const float2*)h, src, dst, summed, deg, n_edges);

    int gm_blocks = (N_NODES + 31) / 32;              // 32 rows per block
    sage_gemm_kernel<<<gm_blocks, 256, 0, stream>>>(
        h, summed, deg, W, bias, out);
}